// GraphConvEncode_84507776516188
// MI455X (gfx1250) — compile-verified
//
#include <hip/hip_runtime.h>
#include <math.h>

// ---------------------------------------------------------------------------
// GVP-GNN autoencoder for gfx1250 (MI455X).
//   N = 4096 nodes, E = 122880 edges, fp32 throughout.
//   All GEMMs on V_WMMA_F32_16X16X4_F32 (fp32 precision preserved).
//   Weights pre-packed into WMMA fragment order -> branch-free inner loops:
//     2x ds_load_b64 (A) + 1x coalesced global_load_b64 (B) + 2x v_wmma.
// ---------------------------------------------------------------------------

#define N_NODES  4096
#define N_EDGES  122880
#define FLAT_S   12800   // 128*100 scalars per graph
#define FLAT_V   6144    // 128*48 vector comps per graph
#define FLAT_K   18944

typedef __attribute__((ext_vector_type(2))) float v2f;
typedef __attribute__((ext_vector_type(8))) float v8f;

struct GvpW  { const v2f *bp; const float *wsb, *wh, *wv; };
struct ConvW { GvpW ff0, ff1; const float *ln0b, *ln0g, *ln1b, *ln1g;
               GvpW msg0, msg1, msg2; };

__device__ __forceinline__ float sigmoidf_(float x) { return 1.f / (1.f + __expf(-x)); }

// ---------------------------------------------------------------------------
// Pack W[K][NO] into fragment order with zero padding:
//   out[((nt*kchunks)+kc)*32 + lane] = {W[kc*4+2*hi][nt*16+lo], W[kc*4+2*hi+1][..]}
// ---------------------------------------------------------------------------
__global__ void pack_b_kernel(const float* __restrict__ W, int K, int NO,
                              int kchunks, long total, v2f* __restrict__ out) {
  long t = (long)blockIdx.x * blockDim.x + threadIdx.x;
  if (t >= total) return;
  const int lane = (int)(t & 31);
  const long rest = t >> 5;
  const int kc = (int)(rest % kchunks);
  const int nt = (int)(rest / kchunks);
  const int lo = lane & 15, hi = lane >> 4;
  const int ka = kc * 4 + (hi << 1);
  const int c  = nt * 16 + lo;
  v2f o;
  o.x = (c < NO && ka     < K) ? W[(size_t)(ka)     * NO + c] : 0.f;
  o.y = (c < NO && ka + 1 < K) ? W[(size_t)(ka + 1) * NO + c] : 0.f;
  out[t] = o;
}

// ---------------------------------------------------------------------------
// Wave-level GEMM with packed B:
//   lsB[32][ldo](cols<NO) = act(lsA[32][lda](cols<K, zero-padded to K4) @ W + b)
// ---------------------------------------------------------------------------
__device__ __forceinline__ void wave_gemm(const v2f* __restrict__ Bp,
                                          const float* __restrict__ bias,
                                          float* lsA, int lda, int K,
                                          float* lsB, int ldo, int NO,
                                          bool relu) {
  const int lane = threadIdx.x & 31;
  const int lo   = lane & 15;
  const int hi   = lane >> 4;
  const int Kp   = (K + 3) & ~3;
  const int pad  = Kp - K;
  const int kchunks = Kp >> 2;
  if (pad) {
    for (int t = lane; t < 32 * pad; t += 32)
      lsA[(t / pad) * lda + K + (t % pad)] = 0.f;
  }
  __syncthreads();
  const int ntiles = (NO + 15) >> 4;
  const float* a0p = lsA + lo * lda + (hi << 1);
  const float* a1p = a0p + 16 * lda;
  for (int nt = 0; nt < ntiles; ++nt) {
    const v2f* pb = Bp + ((size_t)nt * kchunks) * 32 + lane;
    __builtin_prefetch((const void*)pb, 0, 0);
    v8f acc0 = {}; v8f acc1 = {};
    for (int kc = 0; kc < kchunks; ++kc) {
      const v2f a0 = *(const v2f*)(a0p + (kc << 2));
      const v2f a1 = *(const v2f*)(a1p + (kc << 2));
      const v2f bf = *pb;
      pb += 32;
      acc0 = __builtin_amdgcn_wmma_f32_16x16x4_f32(false, a0, false, bf, (short)0, acc0, false, false);
      acc1 = __builtin_amdgcn_wmma_f32_16x16x4_f32(false, a1, false, bf, (short)0, acc1, false, false);
    }
    const int c = (nt << 4) + lo;
    if (c < NO) {
      const float bv = bias[c];
      #pragma unroll
      for (int r = 0; r < 8; ++r) {
        const int row = r + (hi << 3);
        float x0 = acc0[r] + bv;
        float x1 = acc1[r] + bv;
        if (relu) { x0 = fmaxf(x0, 0.f); x1 = fmaxf(x1, 0.f); }
        lsB[row * ldo + c]        = x0;
        lsB[(16 + row) * ldo + c] = x1;
      }
    }
  }
  __syncthreads();
}

// vh = mv @ wh (per coord), vn -> lsA[soff..soff+H)
__device__ __forceinline__ void gvp_vec_pre(const float* __restrict__ wh,
                                            int VI, int H,
                                            const float* lsMV, int ldv,
                                            float* lsVH, int ldh,
                                            float* lsA, int lda, int soff) {
  const int lane = threadIdx.x & 31;
  const float* mv = lsMV + lane * ldv;
  float* vhr = lsVH + lane * ldh;
  float* ar  = lsA + lane * lda;
  for (int h = 0; h < H; ++h) {
    float ax = 0.f, ay = 0.f, az = 0.f;
    for (int i = 0; i < VI; ++i) {
      const float w = wh[i * H + h];        // uniform -> scalar load
      ax += mv[i * 3 + 0] * w;
      ay += mv[i * 3 + 1] * w;
      az += mv[i * 3 + 2] * w;
    }
    vhr[h * 3 + 0] = ax; vhr[h * 3 + 1] = ay; vhr[h * 3 + 2] = az;
    ar[soff + h] = sqrtf(fmaxf(ax * ax + ay * ay + az * az, 1e-8f));
  }
}

// vo = vh @ wv (per coord), optional sigmoid-of-norm gating; writes lsMV rows 0..VO-1
__device__ __forceinline__ void gvp_vec_post(const float* __restrict__ wv,
                                             int H, int VO, bool gate,
                                             const float* lsVH, int ldh,
                                             float* lsMV, int ldv) {
  const int lane = threadIdx.x & 31;
  const float* vhr = lsVH + lane * ldh;
  float* out = lsMV + lane * ldv;
  for (int vv = 0; vv < VO; ++vv) {
    float ax = 0.f, ay = 0.f, az = 0.f;
    for (int h = 0; h < H; ++h) {
      const float w = wv[h * VO + vv];
      ax += vhr[h * 3 + 0] * w;
      ay += vhr[h * 3 + 1] * w;
      az += vhr[h * 3 + 2] * w;
    }
    if (gate) {
      const float g = sigmoidf_(sqrtf(fmaxf(ax * ax + ay * ay + az * az, 1e-8f)));
      ax *= g; ay *= g; az *= g;
    }
    out[vv * 3 + 0] = ax; out[vv * 3 + 1] = ay; out[vv * 3 + 2] = az;
  }
}

// ---------------------------------------------------------------------------
// Small utility kernels
// ---------------------------------------------------------------------------
__global__ void zero_kernel(float* p, long n) {
  long i = (long)blockIdx.x * blockDim.x + threadIdx.x;
  if (i < n) p[i] = 0.f;
}

__global__ void count_kernel(const int* __restrict__ edst, float* cnt, int E) {
  int e = blockIdx.x * blockDim.x + threadIdx.x;
  if (e < E) atomicAdd(&cnt[edst[e]], 1.f);
}

// LayerNorm on scalars + vector RMS normalization, one thread per row
__global__ void ln_kernel(float* sA, float* vA, const float* g, const float* bb,
                          int n, int SD, int VD) {
  int idx = blockIdx.x * blockDim.x + threadIdx.x;
  if (idx >= n) return;
  float* sp = sA + (size_t)idx * SD;
  float mu = 0.f;
  for (int k = 0; k < SD; ++k) mu += sp[k];
  mu /= SD;
  float var = 0.f;
  for (int k = 0; k < SD; ++k) { float d = sp[k] - mu; var += d * d; }
  var /= SD;
  const float inv = rsqrtf(var + 1e-5f);
  for (int k = 0; k < SD; ++k) sp[k] = (sp[k] - mu) * inv * g[k] + bb[k];
  float* vp = vA + (size_t)idx * VD * 3;
  float m = 0.f;
  for (int k = 0; k < VD; ++k) {
    float a = vp[k*3], b = vp[k*3+1], c = vp[k*3+2];
    m += fmaxf(a*a + b*b + c*c, 1e-8f);
  }
  m /= VD;
  const float sc = rsqrtf(m);
  for (int t = 0; t < VD * 3; ++t) vp[t] *= sc;
}

// ---------------------------------------------------------------------------
// Input embeddings (tiny GVPs, per-thread VALU)
// ---------------------------------------------------------------------------
__global__ void node_embed_kernel(const float* __restrict__ hVs, const float* __restrict__ hVv,
                                  const float* __restrict__ wh, const float* __restrict__ wsb,
                                  const float* __restrict__ wsw, const float* __restrict__ wv,
                                  float* s, float* v, int N) {
  int n = blockIdx.x * blockDim.x + threadIdx.x;
  if (n >= N) return;
  const float* si = hVs + (size_t)n * 6;
  const float* vi = hVv + (size_t)n * 9;
  float vh[16][3];
  #pragma unroll
  for (int h = 0; h < 16; ++h) {
    float a = 0.f, b = 0.f, c = 0.f;
    #pragma unroll
    for (int i = 0; i < 3; ++i) {
      const float w = wh[i * 16 + h];
      a += vi[i*3] * w; b += vi[i*3+1] * w; c += vi[i*3+2] * w;
    }
    vh[h][0] = a; vh[h][1] = b; vh[h][2] = c;
  }
  float sin22[22];
  #pragma unroll
  for (int k = 0; k < 6; ++k) sin22[k] = si[k];
  #pragma unroll
  for (int h = 0; h < 16; ++h)
    sin22[6 + h] = sqrtf(fmaxf(vh[h][0]*vh[h][0] + vh[h][1]*vh[h][1] + vh[h][2]*vh[h][2], 1e-8f));
  for (int c = 0; c < 100; ++c) {
    float acc = wsb[c];
    #pragma unroll
    for (int k = 0; k < 22; ++k) acc += sin22[k] * wsw[k * 100 + c];
    s[(size_t)n * 100 + c] = acc;
  }
  #pragma unroll
  for (int vv = 0; vv < 16; ++vv) {
    float a = 0.f, b = 0.f, c = 0.f;
    #pragma unroll
    for (int h = 0; h < 16; ++h) {
      const float w = wv[h * 16 + vv];
      a += vh[h][0] * w; b += vh[h][1] * w; c += vh[h][2] * w;
    }
    v[(size_t)n * 48 + vv*3]   = a;
    v[(size_t)n * 48 + vv*3+1] = b;
    v[(size_t)n * 48 + vv*3+2] = c;
  }
}

__global__ void edge_embed_kernel(const float* __restrict__ hEs, const float* __restrict__ hEv,
                                  const float* __restrict__ wh, const float* __restrict__ wsb,
                                  const float* __restrict__ wsw, const float* __restrict__ wv,
                                  float* es, float* ev, int E) {
  int e = blockIdx.x * blockDim.x + threadIdx.x;
  if (e >= E) return;
  const float* x = hEs + (size_t)e * 32;
  const float w0 = wh[0];
  const float a = hEv[e*3] * w0, b = hEv[e*3+1] * w0, c = hEv[e*3+2] * w0;
  const float vn = sqrtf(fmaxf(a*a + b*b + c*c, 1e-8f));
  for (int o = 0; o < 32; ++o) {
    float acc = wsb[o];
    for (int k = 0; k < 32; ++k) acc += x[k] * wsw[k * 32 + o];
    acc += vn * wsw[32 * 32 + o];
    es[(size_t)e * 32 + o] = acc;
  }
  const float wv0 = wv[0];
  ev[e*3]   = a * wv0;
  ev[e*3+1] = b * wv0;
  ev[e*3+2] = c * wv0;
}

// ---------------------------------------------------------------------------
// Per-edge message GVP chain + scatter. 1 wave == 32 edges.
// ---------------------------------------------------------------------------
#define MSG_LDA 268
#define MSG_LDO 104
#define MSG_LDV 99
#define MSG_SHM (32u * (MSG_LDA + MSG_LDO + MSG_LDV + MSG_LDV) * 4u)

__global__ void __launch_bounds__(32) msg_scatter_kernel(
    const float* __restrict__ s, const float* __restrict__ v,
    const int* __restrict__ esrc, const int* __restrict__ edst,
    const float* __restrict__ es, const float* __restrict__ ev,
    float* __restrict__ ds, float* __restrict__ dv, ConvW W) {
  extern __shared__ float sm[];
  float* lsA  = sm;                       // 32 x 268 scalar GEMM input
  float* lsB  = lsA + 32 * MSG_LDA;       // 32 x 104 GEMM output
  float* lsMV = lsB + 32 * MSG_LDO;       // 32 x 33 x 3 vector channels
  float* lsVH = lsMV + 32 * MSG_LDV;      // 32 x 33 x 3 hidden vectors
  const int lane = threadIdx.x & 31;
  const int e = blockIdx.x * 32 + lane;
  const int j = esrc[e];
  const int i = edst[e];
  float* arow  = lsA + lane * MSG_LDA;
  float* mvrow = lsMV + lane * MSG_LDV;
  // ms = [s[j](100), es(32), s[i](100)] ; mv = [v[j](16), ev(1), v[i](16)]
  {
    const float4* sj = (const float4*)(s + (size_t)j * 100);
    const float4* se = (const float4*)(es + (size_t)e * 32);
    const float4* si = (const float4*)(s + (size_t)i * 100);
    float4* a4 = (float4*)arow;
    #pragma unroll 5
    for (int c = 0; c < 25; ++c) a4[c] = sj[c];
    #pragma unroll
    for (int c = 0; c < 8; ++c)  a4[25 + c] = se[c];
    #pragma unroll 5
    for (int c = 0; c < 25; ++c) a4[33 + c] = si[c];
    const float4* vj = (const float4*)(v + (size_t)j * 48);
    float4* m4 = (float4*)mvrow;
    #pragma unroll
    for (int c = 0; c < 12; ++c) m4[c] = vj[c];
    mvrow[48] = ev[e*3]; mvrow[49] = ev[e*3+1]; mvrow[50] = ev[e*3+2];
    for (int c = 0; c < 48; ++c) mvrow[51 + c] = v[(size_t)i * 48 + c];
  }
  // msg[0]: 232s/33v -> 100s/16v, relu + gate
  gvp_vec_pre (W.msg0.wh, 33, 33, lsMV, MSG_LDV, lsVH, MSG_LDV, lsA, MSG_LDA, 232);
  wave_gemm   (W.msg0.bp, W.msg0.wsb, lsA, MSG_LDA, 265, lsB, MSG_LDO, 100, true);
  gvp_vec_post(W.msg0.wv, 33, 16, true, lsVH, MSG_LDV, lsMV, MSG_LDV);
  {
    const float4* b4 = (const float4*)(lsB + lane * MSG_LDO);
    float4* a4 = (float4*)arow;
    #pragma unroll 5
    for (int c = 0; c < 25; ++c) a4[c] = b4[c];
  }
  __syncthreads();
  // msg[1]: 100s/16v -> 100s/16v, relu + gate
  gvp_vec_pre (W.msg1.wh, 16, 16, lsMV, MSG_LDV, lsVH, MSG_LDV, lsA, MSG_LDA, 100);
  wave_gemm   (W.msg1.bp, W.msg1.wsb, lsA, MSG_LDA, 116, lsB, MSG_LDO, 100, true);
  gvp_vec_post(W.msg1.wv, 16, 16, true, lsVH, MSG_LDV, lsMV, MSG_LDV);
  {
    const float4* b4 = (const float4*)(lsB + lane * MSG_LDO);
    float4* a4 = (float4*)arow;
    #pragma unroll 5
    for (int c = 0; c < 25; ++c) a4[c] = b4[c];
  }
  __syncthreads();
  // msg[2]: no activations
  gvp_vec_pre (W.msg2.wh, 16, 16, lsMV, MSG_LDV, lsVH, MSG_LDV, lsA, MSG_LDA, 100);
  wave_gemm   (W.msg2.bp, W.msg2.wsb, lsA, MSG_LDA, 116, lsB, MSG_LDO, 100, false);
  gvp_vec_post(W.msg2.wv, 16, 16, false, lsVH, MSG_LDV, lsMV, MSG_LDV);
  // scatter (mean applied later with cnt)
  for (int c = 0; c < 100; ++c) atomicAdd(&ds[(size_t)i * 100 + c], lsB[lane * MSG_LDO + c]);
  for (int c = 0; c < 48;  ++c) atomicAdd(&dv[(size_t)i * 48 + c], mvrow[c]);
}

// ---------------------------------------------------------------------------
// Node update: residual + LN0 + feed-forward GVPs (WMMA) + residual + LN1.
// 1 wave == 32 nodes.
// ---------------------------------------------------------------------------
#define NU_LDA 436
#define NU_LDO 400
#define NU_LDV 96
#define NU_SHM (32u * (NU_LDA + NU_LDO + NU_LDV + NU_LDV + 100 + 48) * 4u)

__global__ void __launch_bounds__(32) node_update_kernel(
    float* __restrict__ s, float* __restrict__ v,
    const float* __restrict__ ds, const float* __restrict__ dv,
    const float* __restrict__ cnt, ConvW W) {
  extern __shared__ float sm[];
  float* lsA  = sm;
  float* lsB  = lsA + 32 * NU_LDA;
  float* lsMV = lsB + 32 * NU_LDO;
  float* lsVH = lsMV + 32 * NU_LDV;
  float* lsRS = lsVH + 32 * NU_LDV;   // residual scalars
  float* lsRV = lsRS + 32 * 100;      // residual vectors
  const int lane = threadIdx.x & 31;
  const int n = blockIdx.x * 32 + lane;
  const float inv_c = 1.f / fmaxf(cnt[n], 1.f);
  float* arow  = lsA + lane * NU_LDA;
  float* mvrow = lsMV + lane * NU_LDV;
  float* rs    = lsRS + lane * 100;
  float* rv    = lsRV + lane * 48;
  // s + ds/cnt, LN0
  {
    float mu = 0.f;
    for (int k = 0; k < 100; ++k) {
      float x = s[(size_t)n*100+k] + ds[(size_t)n*100+k] * inv_c;
      rs[k] = x; mu += x;
    }
    mu *= 0.01f;
    float var = 0.f;
    for (int k = 0; k < 100; ++k) { float d = rs[k] - mu; var += d * d; }
    var *= 0.01f;
    const float inv = rsqrtf(var + 1e-5f);
    for (int k = 0; k < 100; ++k) {
      float x = (rs[k] - mu) * inv * W.ln0g[k] + W.ln0b[k];
      rs[k] = x; arow[k] = x;
    }
    float m = 0.f;
    for (int k = 0; k < 16; ++k) {
      float a = v[(size_t)n*48+k*3]   + dv[(size_t)n*48+k*3]   * inv_c;
      float b = v[(size_t)n*48+k*3+1] + dv[(size_t)n*48+k*3+1] * inv_c;
      float c = v[(size_t)n*48+k*3+2] + dv[(size_t)n*48+k*3+2] * inv_c;
      rv[k*3] = a; rv[k*3+1] = b; rv[k*3+2] = c;
      m += fmaxf(a*a + b*b + c*c, 1e-8f);
    }
    m *= (1.f / 16.f);
    const float sc = rsqrtf(m);
    for (int t = 0; t < 48; ++t) { rv[t] *= sc; mvrow[t] = rv[t]; }
  }
  __syncthreads();
  // ff[0]: 100s/16v -> 400s/32v, relu + gate
  gvp_vec_pre (W.ff0.wh, 16, 32, lsMV, NU_LDV, lsVH, NU_LDV, lsA, NU_LDA, 100);
  wave_gemm   (W.ff0.bp, W.ff0.wsb, lsA, NU_LDA, 132, lsB, NU_LDO, 400, true);
  gvp_vec_post(W.ff0.wv, 32, 32, true, lsVH, NU_LDV, lsMV, NU_LDV);
  {
    const float4* b4 = (const float4*)(lsB + lane * NU_LDO);
    float4* a4 = (float4*)arow;
    #pragma unroll 4
    for (int c = 0; c < 100; ++c) a4[c] = b4[c];
  }
  __syncthreads();
  // ff[1]: 400s/32v -> 100s/16v, no act
  gvp_vec_pre (W.ff1.wh, 32, 32, lsMV, NU_LDV, lsVH, NU_LDV, lsA, NU_LDA, 400);
  wave_gemm   (W.ff1.bp, W.ff1.wsb, lsA, NU_LDA, 432, lsB, NU_LDO, 100, false);
  gvp_vec_post(W.ff1.wv, 32, 16, false, lsVH, NU_LDV, lsMV, NU_LDV);
  __syncthreads();
  // residual + LN1 -> global
  {
    float mu = 0.f;
    for (int k = 0; k < 100; ++k) { float x = rs[k] + lsB[lane * NU_LDO + k]; rs[k] = x; mu += x; }
    mu *= 0.01f;
    float var = 0.f;
    for (int k = 0; k < 100; ++k) { float d = rs[k] - mu; var += d * d; }
    var *= 0.01f;
    const float inv = rsqrtf(var + 1e-5f);
    for (int k = 0; k < 100; ++k)
      s[(size_t)n*100+k] = (rs[k] - mu) * inv * W.ln1g[k] + W.ln1b[k];
    float m = 0.f;
    for (int k = 0; k < 16; ++k) {
      float a = rv[k*3]   + mvrow[k*3];
      float b = rv[k*3+1] + mvrow[k*3+1];
      float c = rv[k*3+2] + mvrow[k*3+2];
      rv[k*3] = a; rv[k*3+1] = b; rv[k*3+2] = c;
      m += fmaxf(a*a + b*b + c*c, 1e-8f);
    }
    m *= (1.f / 16.f);
    const float sc = rsqrtf(m);
    for (int t = 0; t < 48; ++t) v[(size_t)n*48+t] = rv[t] * sc;
  }
}

// ---------------------------------------------------------------------------
// Squeeze / unsqueeze MLPs
// ---------------------------------------------------------------------------
__device__ __forceinline__ v2f flat_ld2(const float* s, const float* v, int g, int k) {
  const float* base = (k < FLAT_S) ? (s + (size_t)g * FLAT_S + k)
                                   : (v + (size_t)g * FLAT_V + (k - FLAT_S));
  return *(const v2f*)base;   // pairs never straddle the s/v boundary (k even)
}
__device__ __forceinline__ void flat_st(float* s, float* v, int g, int k, float val) {
  if (k < FLAT_S) s[(size_t)g * FLAT_S + k] = val;
  else            v[(size_t)g * FLAT_V + (k - FLAT_S)] = val;
}

// h1[32][128] = relu(flat[32][18944] @ w + b). grid.x = 8 N-tiles, 1 wave each.
__global__ void __launch_bounds__(32) squeeze_lin0_kernel(
    const float* __restrict__ s, const float* __restrict__ v,
    const v2f* __restrict__ Bp, const float* __restrict__ b, float* __restrict__ h1) {
  const int lane = threadIdx.x & 31;
  const int lo = lane & 15, hi = lane >> 4;
  const int col = blockIdx.x * 16 + lo;     // < 128
  const int kchunks = FLAT_K / 4;
  const v2f* pb = Bp + ((size_t)blockIdx.x * kchunks) * 32 + lane;
  v8f acc0 = {}; v8f acc1 = {};
  for (int kc = 0; kc < kchunks; ++kc) {
    const int ka = (kc << 2) + (hi << 1);
    const v2f a0 = flat_ld2(s, v, lo,      ka);
    const v2f a1 = flat_ld2(s, v, 16 + lo, ka);
    const v2f bf = *pb;
    pb += 32;
    acc0 = __builtin_amdgcn_wmma_f32_16x16x4_f32(false, a0, false, bf, (short)0, acc0, false, false);
    acc1 = __builtin_amdgcn_wmma_f32_16x16x4_f32(false, a1, false, bf, (short)0, acc1, false, false);
  }
  const float bv = b[col];
  #pragma unroll
  for (int r = 0; r < 8; ++r) {
    const int row = r + (hi << 3);
    h1[row * 128 + col]        = fmaxf(acc0[r] + bv, 0.f);
    h1[(16 + row) * 128 + col] = fmaxf(acc1[r] + bv, 0.f);
  }
}

// squeeze lin1+lin2 -> z, then unsqueeze lin0+lin1 -> u2. One lane per graph.
__global__ void __launch_bounds__(32) squeeze_tail_kernel(
    const float* __restrict__ h1,
    const float* __restrict__ w1, const float* __restrict__ b1,
    const float* __restrict__ w2, const float* __restrict__ b2,
    const float* __restrict__ uw0, const float* __restrict__ ub0,
    const float* __restrict__ uw1, const float* __restrict__ ub1,
    float* __restrict__ zout, float* __restrict__ u2) {
  const int g = threadIdx.x & 31;
  float h2[32];
  for (int o = 0; o < 32; ++o) {
    float acc = b1[o];
    for (int k = 0; k < 128; ++k) acc += h1[g * 128 + k] * w1[k * 32 + o];
    h2[o] = fmaxf(acc, 0.f);
  }
  float z[4];
  #pragma unroll
  for (int o = 0; o < 4; ++o) {
    float acc = b2[o];
    for (int k = 0; k < 32; ++k) acc += h2[k] * w2[k * 4 + o];
    z[o] = acc;
    zout[g * 4 + o] = acc;
  }
  float u1[32];
  for (int o = 0; o < 32; ++o) {
    float acc = ub0[o];
    #pragma unroll
    for (int k = 0; k < 4; ++k) acc += z[k] * uw0[k * 32 + o];
    u1[o] = fmaxf(acc, 0.f);
  }
  for (int o = 0; o < 128; ++o) {
    float acc = ub1[o];
    for (int k = 0; k < 32; ++k) acc += u1[k] * uw1[k * 128 + o];
    u2[g * 128 + o] = fmaxf(acc, 0.f);
  }
}

// s/v <- u2[32][128] @ w[128][18944] + b. grid.x = 1184 N-tiles.
__global__ void __launch_bounds__(32) unsqueeze_out_kernel(
    const float* __restrict__ u2, const v2f* __restrict__ Bp, const float* __restrict__ b,
    float* __restrict__ s, float* __restrict__ v) {
  extern __shared__ float sm[];                 // 32 x 128
  const int lane = threadIdx.x & 31;
  const int lo = lane & 15, hi = lane >> 4;
  for (int t = lane; t < 32 * 32; t += 32) ((float4*)sm)[t] = ((const float4*)u2)[t];
  __syncthreads();
  const int col = blockIdx.x * 16 + lo;         // < 18944
  const int kchunks = 128 / 4;
  const v2f* pb = Bp + ((size_t)blockIdx.x * kchunks) * 32 + lane;
  __builtin_prefetch((const void*)pb, 0, 0);
  const float* a0p = sm + lo * 128 + (hi << 1);
  const float* a1p = a0p + 16 * 128;
  v8f acc0 = {}; v8f acc1 = {};
  for (int kc = 0; kc < kchunks; ++kc) {
    const v2f a0 = *(const v2f*)(a0p + (kc << 2));
    const v2f a1 = *(const v2f*)(a1p + (kc << 2));
    const v2f bf = *pb;
    pb += 32;
    acc0 = __builtin_amdgcn_wmma_f32_16x16x4_f32(false, a0, false, bf, (short)0, acc0, false, false);
    acc1 = __builtin_amdgcn_wmma_f32_16x16x4_f32(false, a1, false, bf, (short)0, acc1, false, false);
  }
  const float bv = b[col];
  #pragma unroll
  for (int r = 0; r < 8; ++r) {
    const int row = r + (hi << 3);
    flat_st(s, v, row,      col, acc0[r] + bv);
    flat_st(s, v, 16 + row, col, acc1[r] + bv);
  }
}

// ---------------------------------------------------------------------------
// Output head: pred[n][3] = [s(100), vn(16)] @ wsw + b
// ---------------------------------------------------------------------------
__global__ void wout_kernel(const float* __restrict__ s, const float* __restrict__ v,
                            const float* __restrict__ wh, const float* __restrict__ wsb,
                            const float* __restrict__ wsw, float* __restrict__ pred, int N) {
  int n = blockIdx.x * blockDim.x + threadIdx.x;
  if (n >= N) return;
  float vn[16];
  #pragma unroll
  for (int h = 0; h < 16; ++h) {
    float a = 0.f, b = 0.f, c = 0.f;
    for (int i = 0; i < 16; ++i) {
      const float w = wh[i * 16 + h];
      a += v[(size_t)n*48 + i*3]   * w;
      b += v[(size_t)n*48 + i*3+1] * w;
      c += v[(size_t)n*48 + i*3+2] * w;
    }
    vn[h] = sqrtf(fmaxf(a*a + b*b + c*c, 1e-8f));
  }
  #pragma unroll
  for (int c = 0; c < 3; ++c) {
    float acc = wsb[c];
    for (int k = 0; k < 100; ++k) acc += s[(size_t)n*100+k] * wsw[k * 3 + c];
    #pragma unroll
    for (int h = 0; h < 16; ++h) acc += vn[h] * wsw[(100 + h) * 3 + c];
    pred[(size_t)n * 3 + c] = acc;
  }
}

// ---------------------------------------------------------------------------
// Host launcher
// ---------------------------------------------------------------------------
extern "C" void kernel_launch(void* const* d_in, const int* in_sizes, int n_in,
                              void* d_out, int out_size, void* d_ws, size_t ws_size,
                              hipStream_t stream) {
  (void)in_sizes; (void)n_in; (void)out_size; (void)ws_size;
  const int N = N_NODES, E = N_EDGES;
  const float* hVs = (const float*)d_in[0];
  const float* hVv = (const float*)d_in[1];
  const int*   eix = (const int*)d_in[2];
  const float* hEs = (const float*)d_in[3];
  const float* hEv = (const float*)d_in[4];
  const int* esrc = eix;
  const int* edst = eix + E;

  // params pytree, jax-flattened (dict keys sorted, lists in order):
  // W_e, W_out, W_v, dec[3], enc[3], squeeze[3], unsqueeze[3]
  int p = 5;
  auto F = [&](void) { return (const float*)d_in[p++]; };
  const float *We_wh = F(), *We_wsb = F(), *We_wsw = F(), *We_wv = F();
  const float *We_lnb = F(), *We_lng = F();
  const float *Wo_wh = F(), *Wo_wsb = F(), *Wo_wsw = F();
  const float *Wv_wh = F(), *Wv_wsb = F(), *Wv_wsw = F(), *Wv_wv = F();
  const float *Wv_lnb = F(), *Wv_lng = F();

  struct RawGvp { const float *wh, *wsb, *wsw, *wv; };
  struct RawConv { RawGvp ff0, ff1; const float *ln0b, *ln0g, *ln1b, *ln1g;
                   RawGvp msg0, msg1, msg2; };
  RawConv rdec[3], renc[3];
  auto readGvp = [&](RawGvp& g) { g.wh = F(); g.wsb = F(); g.wsw = F(); g.wv = F(); };
  auto readConv = [&](RawConv& c) {
    readGvp(c.ff0); readGvp(c.ff1);
    c.ln0b = F(); c.ln0g = F(); c.ln1b = F(); c.ln1g = F();
    readGvp(c.msg0); readGvp(c.msg1); readGvp(c.msg2);
  };
  for (int l = 0; l < 3; ++l) readConv(rdec[l]);
  for (int l = 0; l < 3; ++l) readConv(renc[l]);
  const float *sq_b0 = F(), *sq_w0 = F(), *sq_b1 = F(), *sq_w1 = F(), *sq_b2 = F(), *sq_w2 = F();
  const float *un_b0 = F(), *un_w0 = F(), *un_b1 = F(), *un_w1 = F(), *un_b2 = F(), *un_w2 = F();

  // workspace layout (floats)
  float* ws = (float*)d_ws;
  float* s   = ws;                       // N*100
  float* v   = s + (size_t)N * 100;      // N*48
  float* es  = v + (size_t)N * 48;       // E*32
  float* ev  = es + (size_t)E * 32;      // E*3
  float* ds  = ev + (size_t)E * 3;       // N*100
  float* dv  = ds + (size_t)N * 100;     // N*48
  float* cnt = dv + (size_t)N * 48;      // N
  float* h1  = cnt + N;                  // 32*128
  float* u2  = h1 + 32 * 128;            // 32*128
  v2f* packArena = (v2f*)(u2 + 32 * 128);
  v2f* packCur = packArena;
  float* pred = (float*)d_out;           // N*3
  float* zout = pred + (size_t)N * 3;    // 32*4

  // ---- pre-pack all GEMM weights into WMMA fragment order
  auto packB = [&](const float* W, int K, int NO) -> const v2f* {
    const int Kp = (K + 3) & ~3, kch = Kp >> 2, ntl = (NO + 15) >> 4;
    const long total = (long)ntl * kch * 32;
    v2f* dst = packCur; packCur += total;
    pack_b_kernel<<<(int)((total + 255) / 256), 256, 0, stream>>>(W, K, NO, kch, total, dst);
    return dst;
  };
  ConvW dec[3], enc[3];
  auto buildConv = [&](const RawConv& r, ConvW& c, bool /*unused*/) {
    c.ff0  = { packB(r.ff0.wsw, 132, 400),  r.ff0.wsb,  r.ff0.wh,  r.ff0.wv };
    c.ff1  = { packB(r.ff1.wsw, 432, 100),  r.ff1.wsb,  r.ff1.wh,  r.ff1.wv };
    c.ln0b = r.ln0b; c.ln0g = r.ln0g; c.ln1b = r.ln1b; c.ln1g = r.ln1g;
    c.msg0 = { packB(r.msg0.wsw, 265, 100), r.msg0.wsb, r.msg0.wh, r.msg0.wv };
    c.msg1 = { packB(r.msg1.wsw, 116, 100), r.msg1.wsb, r.msg1.wh, r.msg1.wv };
    c.msg2 = { packB(r.msg2.wsw, 116, 100), r.msg2.wsb, r.msg2.wh, r.msg2.wv };
  };
  for (int l = 0; l < 3; ++l) buildConv(rdec[l], dec[l], false);
  for (int l = 0; l < 3; ++l) buildConv(renc[l], enc[l], false);
  const v2f* sq_p0 = packB(sq_w0, FLAT_K, 128);
  const v2f* un_p2 = packB(un_w2, 128, FLAT_K);

  // ---- embeddings
  node_embed_kernel<<<N / 64, 64, 0, stream>>>(hVs, hVv, Wv_wh, Wv_wsb, Wv_wsw, Wv_wv, s, v, N);
  ln_kernel<<<(N + 255) / 256, 256, 0, stream>>>(s, v, Wv_lng, Wv_lnb, N, 100, 16);
  edge_embed_kernel<<<E / 256, 256, 0, stream>>>(hEs, hEv, We_wh, We_wsb, We_wsw, We_wv, es, ev, E);
  ln_kernel<<<(E + 255) / 256, 256, 0, stream>>>(es, ev, We_lng, We_lnb, E, 32, 1);

  // ---- degree counts
  zero_kernel<<<(N + 255) / 256, 256, 0, stream>>>(cnt, N);
  count_kernel<<<E / 256, 256, 0, stream>>>(edst, cnt, E);

  const long accN = (long)N * 148;
  // ---- encoder
  for (int l = 0; l < 3; ++l) {
    zero_kernel<<<(int)((accN + 255) / 256), 256, 0, stream>>>(ds, accN);
    msg_scatter_kernel<<<E / 32, 32, MSG_SHM, stream>>>(s, v, esrc, edst, es, ev, ds, dv, enc[l]);
    node_update_kernel<<<N / 32, 32, NU_SHM, stream>>>(s, v, ds, dv, cnt, enc[l]);
  }
  // ---- squeeze -> z -> unsqueeze
  squeeze_lin0_kernel<<<8, 32, 0, stream>>>(s, v, sq_p0, sq_b0, h1);
  squeeze_tail_kernel<<<1, 32, 0, stream>>>(h1, sq_w1, sq_b1, sq_w2, sq_b2,
                                            un_w0, un_b0, un_w1, un_b1, zout, u2);
  unsqueeze_out_kernel<<<FLAT_K / 16, 32, 32 * 128 * 4, stream>>>(u2, un_p2, un_b2, s, v);
  // ---- decoder
  for (int l = 0; l < 3; ++l) {
    zero_kernel<<<(int)((accN + 255) / 256), 256, 0, stream>>>(ds, accN);
    msg_scatter_kernel<<<E / 32, 32, MSG_SHM, stream>>>(s, v, esrc, edst, es, ev, ds, dv, dec[l]);
    node_update_kernel<<<N / 32, 32, NU_SHM, stream>>>(s, v, ds, dv, cnt, dec[l]);
  }
  // ---- output head
  wout_kernel<<<N / 64, 64, 0, stream>>>(s, v, Wo_wh, Wo_wsb, Wo_wsw, pred, N);
}